// ResNet_28020366639550
// MI455X (gfx1250) — compile-verified
//
#include <hip/hip_runtime.h>
#include <math.h>

typedef __attribute__((ext_vector_type(2))) float v2f;
typedef __attribute__((ext_vector_type(8))) float v8f;

#define N_ROWS 1500000
#define C 32
#define KOFF 8
#define BN_EPS 1e-5f

// ---------------------------------------------------------------------------
// Gather + GEMM conv using V_WMMA_F32_16X16X4_F32 (CDNA5 fp32 matrix op).
// Each wave computes a 32-row x 32-col output tile (two 16-row M-tiles that
// share B operands, four independent accumulator chains):
//   out[32,32] = sum_k gather(src, idx[k]) @ W[k]
//
// The cin contraction is processed in a PERMUTED chunk order (applied to both
// A columns and B rows, so the sum is unchanged):
//   chunk j, lane-half h, reg r  ->  cin = 16h + 2j + r
// This makes each lane's 16 A-elements contiguous (channels 16h..16h+15), so
// the per-offset gather is 4 x global_load_b128 per row, hoisted ahead of the
// j-loop, instead of 8 interleaved 8B loads.
//
// Hardware layouts (wave32):
//   A (16x4 f32): lanes 0-15 hold M=0..15; reg r, lane-half h -> chunk K=2h+r.
//   B (4x16 f32): reg r, lane-half h -> chunk K=2h+r, N=lane&15.
//   D (16x16 f32): reg v, lane-half h -> M=v+8h, N=lane&15 (per column half).
// Weights are staged into LDS pre-swizzled into the per-lane B order
// (including the cin permutation) so each j-step is one fused pair of
// ds_load_b64 feeding 4 WMMAs.
//
// The k-loop is force-unrolled so the scheduler can pipeline the next
// offset's gathers above the current offset's WMMA chain (the random gather
// latency is the critical path; occupancy alone shouldn't have to hide it).
//
// APPLY_BN: apply y = relu(v*scale[c]+shift[c]) to gathered rows on the fly
// (fuses BN1+ReLU into conv2's gather, saving a full 192MB pass). Per-lane
// scale/shift (16 contiguous channels) are hoisted into registers.
// Also accumulates per-channel sum / sum-of-squares for the following BN.
// ---------------------------------------------------------------------------
template <bool APPLY_BN>
__global__ __launch_bounds__(256) void conv_wmma_kernel(
    const float* __restrict__ src,       // [N, 32]
    const int* __restrict__ nidx,        // [8, N], value N -> zero row
    const float* __restrict__ Wg,        // [8, 32, 32]
    const float* __restrict__ bn_scale,  // [32] (used iff APPLY_BN)
    const float* __restrict__ bn_shift,  // [32]
    float* __restrict__ dst,             // [N, 32]
    float* __restrict__ gsum,            // [32]
    float* __restrict__ gsumsq)          // [32]
{
  __shared__ float lW[KOFF * 8 * 2 * 64];  // 8192 floats = 32KB, swizzled B
  __shared__ float lscale[C];
  __shared__ float lshift[C];
  __shared__ float sred[64];               // [0:32) sum, [32:64) sumsq

  const int tid = threadIdx.x;

  // Stage weights swizzled into per-lane WMMA-B order (with cin permutation):
  // lW[((k*8+j)*2+half)*64 + lane*2 + r]
  //   = W[k][16*(lane>>4) + 2*j + r][16*half + (lane&15)]
  // Batched through registers so the compiler clauses the global loads
  // (one wait) instead of a serialized load->wait->ds_store chain.
  {
    float wtmp[32];
#pragma unroll
    for (int it = 0; it < 32; ++it) {
      const int t = tid + it * 256;
      const int r    = t & 1;
      const int lane = (t >> 1) & 31;
      const int half = (t >> 6) & 1;
      const int j    = (t >> 7) & 7;
      const int k    = t >> 10;
      const int kc   = 16 * (lane >> 4) + 2 * j + r;
      const int cout = 16 * half + (lane & 15);
      wtmp[it] = Wg[k * 1024 + kc * 32 + cout];
    }
#pragma unroll
    for (int it = 0; it < 32; ++it) {
      lW[tid + it * 256] = wtmp[it];
    }
  }
  if (APPLY_BN && tid < C) {
    lscale[tid] = bn_scale[tid];
    lshift[tid] = bn_shift[tid];
  }
  if (tid < 64) sred[tid] = 0.0f;
  __syncthreads();

  const int lane = tid & 31;
  const int wave = tid >> 5;
  const int h    = lane >> 4;   // lane half: selects cin block (A/B), row group (D)
  const int m    = lane & 15;   // A: row-in-tile; D: column-in-half
  const long tile = (long)blockIdx.x * 8 + wave;
  const long row0 = tile * 32;
  const bool active = row0 < (long)N_ROWS;  // N % 32 == 0 -> tiles always full

  v8f acc00 = {0.f, 0.f, 0.f, 0.f, 0.f, 0.f, 0.f, 0.f};  // M-tile 0, cols 0-15
  v8f acc01 = {0.f, 0.f, 0.f, 0.f, 0.f, 0.f, 0.f, 0.f};  // M-tile 0, cols 16-31
  v8f acc10 = {0.f, 0.f, 0.f, 0.f, 0.f, 0.f, 0.f, 0.f};  // M-tile 1, cols 0-15
  v8f acc11 = {0.f, 0.f, 0.f, 0.f, 0.f, 0.f, 0.f, 0.f};  // M-tile 1, cols 16-31

  if (active) {
    // BN scale/shift for this lane's 16 contiguous channels, hoisted.
    float bs[16], bf[16];
    if (APPLY_BN) {
#pragma unroll
      for (int i = 0; i < 16; ++i) {
        bs[i] = lscale[16 * h + i];
        bf[i] = lshift[16 * h + i];
      }
    }

    const long grow0 = row0 + m;        // gather row for M-tile 0
    const long grow1 = row0 + 16 + m;   // gather row for M-tile 1
#pragma unroll
    for (int k = 0; k < KOFF; ++k) {
      const long kbase = (long)k * N_ROWS;
      const int g0 = nidx[kbase + grow0];
      const int g1 = nidx[kbase + grow1];
      const bool valid0 = (g0 < N_ROWS);  // g == N -> zero padding row
      const bool valid1 = (g1 < N_ROWS);

      // Gather this lane's 16 contiguous channels of each row: 4 x b128.
      float a0r[16], a1r[16];
#pragma unroll
      for (int i = 0; i < 16; ++i) { a0r[i] = 0.0f; a1r[i] = 0.0f; }
      if (valid0) {
        const float4* q = (const float4*)(src + (long)g0 * C + 16 * h);
#pragma unroll
        for (int i = 0; i < 4; ++i) {
          const float4 t = q[i];
          a0r[4 * i] = t.x; a0r[4 * i + 1] = t.y;
          a0r[4 * i + 2] = t.z; a0r[4 * i + 3] = t.w;
        }
      }
      if (valid1) {
        const float4* q = (const float4*)(src + (long)g1 * C + 16 * h);
#pragma unroll
        for (int i = 0; i < 4; ++i) {
          const float4 t = q[i];
          a1r[4 * i] = t.x; a1r[4 * i + 1] = t.y;
          a1r[4 * i + 2] = t.z; a1r[4 * i + 3] = t.w;
        }
      }
      if (APPLY_BN) {
#pragma unroll
        for (int i = 0; i < 16; ++i) {
          a0r[i] = fmaxf(a0r[i] * bs[i] + bf[i], 0.0f);
          a1r[i] = fmaxf(a1r[i] * bs[i] + bf[i], 0.0f);
        }
      }

#pragma unroll
      for (int j = 0; j < 8; ++j) {
        const v2f a0 = {a0r[2 * j], a0r[2 * j + 1]};
        const v2f a1 = {a1r[2 * j], a1r[2 * j + 1]};
        const int base = ((k * 8 + j) * 2) * 64 + lane * 2;
        const v2f b0 = *(const v2f*)&lW[base];       // column half 0
        const v2f b1 = *(const v2f*)&lW[base + 64];  // column half 1
        acc00 = __builtin_amdgcn_wmma_f32_16x16x4_f32(
            false, a0, false, b0, (short)0, acc00, false, false);
        acc01 = __builtin_amdgcn_wmma_f32_16x16x4_f32(
            false, a0, false, b1, (short)0, acc01, false, false);
        acc10 = __builtin_amdgcn_wmma_f32_16x16x4_f32(
            false, a1, false, b0, (short)0, acc10, false, false);
        acc11 = __builtin_amdgcn_wmma_f32_16x16x4_f32(
            false, a1, false, b1, (short)0, acc11, false, false);
      }
    }

    // Store D tiles + per-channel partial stats.
    float s0 = 0.f, q0 = 0.f, s1 = 0.f, q1 = 0.f;
#pragma unroll
    for (int v = 0; v < 8; ++v) {
      const long r0 = row0 + v + 8 * h;        // M-tile 0
      const long r1 = row0 + 16 + v + 8 * h;   // M-tile 1
      const float d00 = acc00[v], d01 = acc01[v];
      const float d10 = acc10[v], d11 = acc11[v];
      dst[r0 * C + m]      = d00;
      dst[r0 * C + 16 + m] = d01;
      dst[r1 * C + m]      = d10;
      dst[r1 * C + 16 + m] = d11;
      s0 += d00 + d10; q0 += d00 * d00 + d10 * d10;   // column m
      s1 += d01 + d11; q1 += d01 * d01 + d11 * d11;   // column 16+m
    }
    atomicAdd(&sred[m], s0);
    atomicAdd(&sred[32 + m], q0);
    atomicAdd(&sred[16 + m], s1);
    atomicAdd(&sred[32 + 16 + m], q1);
  }
  __syncthreads();
  if (tid < 32) {
    atomicAdd(&gsum[tid], sred[tid]);
    atomicAdd(&gsumsq[tid], sred[32 + tid]);
  }
}

// Fold per-channel sums into BN scale/shift: y = v*scale[c] + shift[c]
__global__ void bn_stats_kernel(const float* __restrict__ sum,
                                const float* __restrict__ sumsq,
                                const float* __restrict__ gamma,
                                const float* __restrict__ beta,
                                float* __restrict__ scale,
                                float* __restrict__ shift) {
  const int c = threadIdx.x;
  if (c < C) {
    const float inv_n = 1.0f / (float)N_ROWS;
    const float mean = sum[c] * inv_n;
    const float var = sumsq[c] * inv_n - mean * mean;
    const float s = gamma[c] * rsqrtf(var + BN_EPS);
    scale[c] = s;
    shift[c] = beta[c] - mean * s;
  }
}

__global__ void zero_kernel(float* __restrict__ p, int n) {
  const int i = blockIdx.x * blockDim.x + threadIdx.x;
  if (i < n) p[i] = 0.0f;
}

// out = bn2(out) + x, in place, vectorized float4 (C=32 divisible by 4).
// Grid-stride step (8192*256*4 elements) is a multiple of 32, so each
// thread's channel group is invariant: hoist scale/shift into registers.
__global__ __launch_bounds__(256) void bn_residual_kernel(
    float* __restrict__ out, const float* __restrict__ x,
    const float* __restrict__ scale, const float* __restrict__ shift) {
  const long total = (long)N_ROWS * C / 4;
  const long start = (long)blockIdx.x * blockDim.x + threadIdx.x;
  const int c0 = (int)((start * 4) & 31);
  const float s0 = scale[c0 + 0], s1 = scale[c0 + 1];
  const float s2 = scale[c0 + 2], s3 = scale[c0 + 3];
  const float f0 = shift[c0 + 0], f1 = shift[c0 + 1];
  const float f2 = shift[c0 + 2], f3 = shift[c0 + 3];
  for (long i = start; i < total; i += (long)gridDim.x * blockDim.x) {
    float4 hv = ((const float4*)out)[i];
    const float4 xv = ((const float4*)x)[i];
    hv.x = hv.x * s0 + f0 + xv.x;
    hv.y = hv.y * s1 + f1 + xv.y;
    hv.z = hv.z * s2 + f2 + xv.z;
    hv.w = hv.w * s3 + f3 + xv.w;
    ((float4*)out)[i] = hv;
  }
}

extern "C" void kernel_launch(void* const* d_in, const int* in_sizes, int n_in,
                              void* d_out, int out_size, void* d_ws,
                              size_t ws_size, hipStream_t stream) {
  const float* x   = (const float*)d_in[0];
  const int* nidx  = (const int*)d_in[1];
  const float* W1  = (const float*)d_in[2];
  const float* g1  = (const float*)d_in[3];
  const float* b1  = (const float*)d_in[4];
  const float* W2  = (const float*)d_in[5];
  const float* g2  = (const float*)d_in[6];
  const float* b2  = (const float*)d_in[7];
  float* out = (float*)d_out;

  float* h1    = (float*)d_ws;                  // [N, 32] raw conv1 output
  float* stats = h1 + (size_t)N_ROWS * C;       // 256 floats of BN state
  float* sum1 = stats +   0; float* sumsq1 = stats +  32;
  float* scl1 = stats +  64; float* shf1   = stats +  96;
  float* sum2 = stats + 128; float* sumsq2 = stats + 160;
  float* scl2 = stats + 192; float* shf2   = stats + 224;

  const int n_tiles = N_ROWS / 32;              // 46875 (exact)
  const int conv_blocks = (n_tiles + 7) / 8;    // 8 waves (32-row tiles)/block

  // Stats accumulators must be zero every call (graph-replayed).
  zero_kernel<<<1, 256, 0, stream>>>(stats, 256);

  // conv1: h1 = sum_k gather(x) @ W1[k]; accumulate BN1 stats.
  conv_wmma_kernel<false><<<conv_blocks, 256, 0, stream>>>(
      x, nidx, W1, nullptr, nullptr, h1, sum1, sumsq1);

  bn_stats_kernel<<<1, 32, 0, stream>>>(sum1, sumsq1, g1, b1, scl1, shf1);

  // conv2 with BN1+ReLU fused into the gather; raw h2 -> d_out; BN2 stats.
  conv_wmma_kernel<true><<<conv_blocks, 256, 0, stream>>>(
      h1, nidx, W2, scl1, shf1, out, sum2, sumsq2);

  bn_stats_kernel<<<1, 32, 0, stream>>>(sum2, sumsq2, g2, b2, scl2, shf2);

  // out = bn2(out) + x
  bn_residual_kernel<<<8192, 256, 0, stream>>>(out, x, scl2, shf2);
}